// BinaryTokenClassificationModel_54150947668678
// MI455X (gfx1250) — compile-verified
//
#include <hip/hip_runtime.h>
#include <hip/hip_bf16.h>

// Problem constants (fixed by the reference harness)
#define BB   8
#define LL   512      // L_SRC + L_TGT
#define LSRC 256
#define HH   768
#define SS   128
#define TT   128
#define NSEG 257      // S+T runs + 1 invalid bucket

#define KC      128   // K-chunk (floats) staged per TDM load
#define NCHUNK  (HH / KC)          // 6
#define ROWPAD  (KC + 4)           // LDS row stride in floats (TDM pad: +4 DW)

typedef __attribute__((ext_vector_type(2))) float v2f;
typedef __attribute__((ext_vector_type(8))) float v8f;
typedef unsigned int u32x4 __attribute__((ext_vector_type(4)));
typedef int i32x4 __attribute__((ext_vector_type(4)));
typedef int i32x8 __attribute__((ext_vector_type(8)));

// ---------------------------------------------------------------------------
// TDM: issue tensor_load_to_lds of a 16 x KC f32 tile (row stride HH floats)
// from global `gaddr` into LDS byte-offset `lds_off`, with LDS padding of
// 4 DWORDs after every 128 DWORDs stored (-> ROWPAD-float LDS rows).
// Descriptor layout per CDNA5 ISA.
// ---------------------------------------------------------------------------
__device__ __forceinline__ void tdm_load_tile(unsigned long long gaddr,
                                              unsigned lds_off) {
    u32x4 g0;
    g0[0] = 1u;                                    // count=1 (valid user D#)
    g0[1] = lds_off;                               // lds_addr (bytes)
    g0[2] = (unsigned)(gaddr & 0xFFFFFFFFull);     // global_addr[31:0]
    g0[3] = (unsigned)((gaddr >> 32) & 0x1FFFFFFull) | (2u << 30); // [56:32]+type=2

    const unsigned td0 = 1u << 30;                 // huge tensor dims: no OOB clamp
    const unsigned td1 = 1u << 30;
    i32x8 g1;
    g1[0] = (int)((2u << 16)        // data_size = 4 bytes
                | (1u << 20)        // pad_enable
                | (6u << 22)        // pad_interval: 2^6 * 8B = 128 DWORDs
                | (3u << 25));      // pad_amount: 4 DWORDs
    g1[1] = (int)((td0 & 0xFFFFu) << 16);                      // tensor_dim0 lo16
    g1[2] = (int)(((td0 >> 16) & 0xFFFFu) | ((td1 & 0xFFFFu) << 16));
    g1[3] = (int)(((td1 >> 16) & 0xFFFFu) | ((unsigned)KC << 16)); // tile_dim0=KC
    g1[4] = (int)16u;                              // tile_dim1=16 rows, tile_dim2=0
    g1[5] = (int)HH;                               // tensor_dim0_stride lo32 = 768
    g1[6] = 0;                                     // stride hi16, dim1_stride lo16
    g1[7] = 0;

    i32x4 z4 = {0, 0, 0, 0};                       // groups 2/3: 2-D tile, unused
#if defined(__clang_major__) && (__clang_major__ >= 23)
    i32x8 z8 = {0, 0, 0, 0, 0, 0, 0, 0};
    __builtin_amdgcn_tensor_load_to_lds(g0, g1, z4, z4, z8, 0);
#else
    __builtin_amdgcn_tensor_load_to_lds(g0, g1, z4, z4, 0);
#endif
}

// ---------------------------------------------------------------------------
// Kernel 1: per-batch run scan (run ids, run starts, counts).
// One block per batch, 512 threads.
// ---------------------------------------------------------------------------
__global__ void __launch_bounds__(LL)
seg_scan_kernel(const int* __restrict__ mask,
                const int* __restrict__ src_ids,
                const int* __restrict__ tgt_ids,
                int* __restrict__ seg,        // [B][L]
                int* __restrict__ runstart,   // [B][256]
                float* __restrict__ counts)   // [B][NSEG]
{
    const int b = blockIdx.x;
    const int l = threadIdx.x;

    __shared__ int sh_wid[LL];
    __shared__ int sh_valid[LL];
    __shared__ int sh_scan[LL];

    const int wid = (l < LSRC) ? src_ids[b * LSRC + l]
                               : tgt_ids[b * LSRC + (l - LSRC)];
    const int valid = (mask[b * LL + l] > 0) && (wid >= 0);
    sh_wid[l] = wid;
    sh_valid[l] = valid;
    __syncthreads();

    int new_run = 0;
    if (valid) {
        if (l == 0) new_run = 1;
        else        new_run = (wid != sh_wid[l - 1]) || !sh_valid[l - 1];
    }
    sh_scan[l] = new_run;
    __syncthreads();

    for (int off = 1; off < LL; off <<= 1) {   // Hillis-Steele inclusive scan
        int v   = sh_scan[l];
        int add = (l >= off) ? sh_scan[l - off] : 0;
        __syncthreads();
        sh_scan[l] = v + add;
        __syncthreads();
    }

    const int run_id = sh_scan[l] - 1;
    const int s = valid ? run_id : (NSEG - 1);
    seg[b * LL + l] = s;

    if (l < NSEG)     counts[b * NSEG + l] = 0.0f;
    if (l < SS + TT)  runstart[b * (SS + TT) + l] = LL;   // sentinel
    __syncthreads();

    if (valid && s < SS + TT) {
        atomicAdd(&counts[b * NSEG + s], 1.0f);   // sum of 1.0f's: order-invariant
        if (new_run) runstart[b * (SS + TT) + s] = l;
    }
}

// ---------------------------------------------------------------------------
// Kernel 2: per-token dual dot products. TDM -> LDS -> V_WMMA_F32_16X16X4_F32.
// One wave (32 lanes) per 16-token tile. A tile (16 x 768) is staged through
// LDS in six 16x128 chunks, double-buffered, overlapping TDM with WMMA.
//   A layout (16x4 f32): lane m=lane&15 -> row M=m; kg=lane>>4 -> K pair.
//   B layout (4x16 f32): col N=lane&15; kg -> K pair. B = [w_src | w_tgt | 0..]
//   D layout (16x16 f32): VGPR i -> row 8*(lane>>4)+i; col = lane&15. Cols 0/1 = dots.
// ---------------------------------------------------------------------------
__global__ void __launch_bounds__(32)
tokdot_wmma_kernel(const float* __restrict__ tok,   // [B][L][H]
                   const float* __restrict__ Wf,    // [2H]
                   float* __restrict__ tokdot)      // [B][L][2]
{
    __shared__ float abuf[2][16 * ROWPAD];

    const int tile = blockIdx.x;             // 0 .. B*L/16-1
    const int b    = tile / (LL / 16);
    const int row0 = (tile % (LL / 16)) * 16;
    const int lane = threadIdx.x;            // wave32
    const int m    = lane & 15;
    const int kg   = lane >> 4;              // K pair {0,1} / {2,3}

    // Tile base in global memory; LDS byte offsets (flat LDS addr low 32 bits).
    const unsigned long long gbase =
        (unsigned long long)(uintptr_t)(tok + (size_t)(b * LL + row0) * HH);
    const unsigned lds_off0 = (unsigned)(uintptr_t)(void*)&abuf[0][0];
    const unsigned lds_off1 = (unsigned)(uintptr_t)(void*)&abuf[1][0];

    const float bscale = (m < 2) ? 1.0f : 0.0f;          // pad B cols 2..15 to 0
    const float* wbase = Wf + ((m == 1) ? HH : 0) + kg * 2;

    // Prime the pipeline: chunk 0 -> buffer 0.
    tdm_load_tile(gbase, lds_off0);

    v8f acc = {};
    for (int c = 0; c < NCHUNK; ++c) {
        if (c + 1 < NCHUNK) {
            // Overlap: kick off next chunk's DMA into the other buffer, then
            // wait until only that just-issued op is outstanding (TENSORcnt is
            // in-order per wave) -> chunk c is resident in LDS.
            tdm_load_tile(gbase + (unsigned long long)(c + 1) * KC * 4,
                          ((c + 1) & 1) ? lds_off1 : lds_off0);
            __builtin_amdgcn_s_wait_tensorcnt(1);
        } else {
            // Last chunk: drain everything.
            __builtin_amdgcn_s_wait_tensorcnt(0);
        }
        __asm__ volatile("" ::: "memory");   // keep LDS reads after the wait

        const float* lrow = &abuf[c & 1][m * ROWPAD + kg * 2];
        const float* wch  = wbase + c * KC;
        __builtin_prefetch(wch + KC, 0, 1);  // global_prefetch_b8 for next W chunk

#pragma unroll 8
        for (int kk = 0; kk < KC; kk += 4) {
            v2f av;
            av.x = lrow[kk];
            av.y = lrow[kk + 1];
            v2f bv;
            bv.x = wch[kk]     * bscale;
            bv.y = wch[kk + 1] * bscale;
            acc = __builtin_amdgcn_wmma_f32_16x16x4_f32(
                      /*neg_a=*/false, av,
                      /*neg_b=*/false, bv,
                      /*c_mod=*/(short)0, acc,
                      /*reuse_a=*/false, /*reuse_b=*/false);
        }
    }

    // Extract columns 0 (src dot) and 1 (tgt dot).
    if (m < 2) {
        const int ch    = m;
        const int rbase = kg * 8;    // lanes 0-15: rows 0..7; lanes 16-31: 8..15
#pragma unroll
        for (int i = 0; i < 8; ++i) {
            const int row = rbase + i;
            tokdot[((size_t)(b * LL) + row0 + row) * 2 + ch] = acc[i];
        }
    }
}

// ---------------------------------------------------------------------------
// Kernel 3: deterministic per-run reduction (one thread per (batch, run)).
// ---------------------------------------------------------------------------
__global__ void __launch_bounds__(256)
pool_reduce_kernel(const float* __restrict__ tokdot,    // [B][L][2]
                   const int* __restrict__ runstart,    // [B][256]
                   const float* __restrict__ counts,    // [B][NSEG]
                   float* __restrict__ a_out,           // [B][S]
                   float* __restrict__ c_out)           // [B][T]
{
    const int idx = blockIdx.x * blockDim.x + threadIdx.x;
    if (idx >= BB * (SS + TT)) return;
    const int b = idx >> 8;
    const int s = idx & 255;

    const float cnt   = counts[b * NSEG + s];
    const int   start = runstart[b * (SS + TT) + s];
    const int   n     = (int)cnt;
    const int   ch    = (s < SS) ? 0 : 1;

    float sum = 0.0f;
    for (int i = 0; i < n; ++i) {
        const int l = start + i;
        sum += tokdot[((size_t)(b * LL) + l) * 2 + ch];
    }
    const float v = sum / fmaxf(cnt, 1.0f);
    if (s < SS) a_out[b * SS + s]        = v;
    else        c_out[b * TT + (s - SS)] = v;
}

// ---------------------------------------------------------------------------
// Kernel 4: logits[b,s,t] = a[b,s] + c[b,t] + bias
// ---------------------------------------------------------------------------
__global__ void __launch_bounds__(256)
logits_kernel(const float* __restrict__ a_out,
              const float* __restrict__ c_out,
              const float* __restrict__ bias,
              float* __restrict__ out)
{
    const int idx = blockIdx.x * blockDim.x + threadIdx.x;
    if (idx >= BB * SS * TT) return;
    const int t = idx & (TT - 1);
    const int s = (idx >> 7) & (SS - 1);
    const int b = idx >> 14;
    out[idx] = a_out[b * SS + s] + c_out[b * TT + t] + bias[0];
}

// ---------------------------------------------------------------------------
extern "C" void kernel_launch(void* const* d_in, const int* in_sizes, int n_in,
                              void* d_out, int out_size, void* d_ws, size_t ws_size,
                              hipStream_t stream) {
    const float* tok_h   = (const float*)d_in[0];   // [B][L][H] f32
    const int*   mask    = (const int*)  d_in[1];   // [B][L]
    const int*   src_ids = (const int*)  d_in[2];   // [B][256]
    const int*   tgt_ids = (const int*)  d_in[3];   // [B][256]
    const float* Wf      = (const float*)d_in[4];   // [2H]
    const float* bias    = (const float*)d_in[5];   // [1]
    float* out = (float*)d_out;                     // [B][S][T]

    char* ws = (char*)d_ws;
    size_t off = 0;
    auto carve = [&](size_t bytes) -> char* {
        char* p = ws + off;
        off = (off + bytes + 255) & ~(size_t)255;
        return p;
    };
    int*   seg      = (int*)  carve((size_t)BB * LL * sizeof(int));
    int*   runstart = (int*)  carve((size_t)BB * (SS + TT) * sizeof(int));
    float* counts   = (float*)carve((size_t)BB * NSEG * sizeof(float));
    float* tokdot   = (float*)carve((size_t)BB * LL * 2 * sizeof(float));
    float* a_out    = (float*)carve((size_t)BB * SS * sizeof(float));
    float* c_out    = (float*)carve((size_t)BB * TT * sizeof(float));
    (void)seg; (void)ws_size; (void)in_sizes; (void)n_in; (void)out_size;

    seg_scan_kernel<<<BB, LL, 0, stream>>>(mask, src_ids, tgt_ids,
                                           seg, runstart, counts);

    tokdot_wmma_kernel<<<BB * (LL / 16), 32, 0, stream>>>(tok_h, Wf, tokdot);

    pool_reduce_kernel<<<(BB * (SS + TT) + 255) / 256, 256, 0, stream>>>(
        tokdot, runstart, counts, a_out, c_out);

    logits_kernel<<<(BB * SS * TT + 255) / 256, 256, 0, stream>>>(
        a_out, c_out, bias, out);
}